// BilinearDecoder_18674517803527
// MI455X (gfx1250) — compile-verified
//
#include <hip/hip_runtime.h>
#include <math.h>

typedef __attribute__((ext_vector_type(2))) float v2f;
typedef __attribute__((ext_vector_type(8))) float v8f;

#define LATENT 128

// ---------------------------------------------------------------------------
// Kernel 1: U = Z @ W  (Z: [N,128] row-major, W: [128,128] row-major)
// One wave32 per 16x16 output tile; K=128 via 32x v_wmma_f32_16x16x4_f32.
// A 16x4 f32 layout : lanes 0-15 -> rows M=lane, VGPR{0,1}=K{0,1};
//                     lanes 16-31 -> rows M=lane-16, VGPR{0,1}=K{2,3}.
// B 4x16 f32 layout : lanes 0-15 -> cols N=lane, VGPR{0,1}=K{0,1};
//                     lanes 16-31 -> cols N=lane-16, VGPR{0,1}=K{2,3}.
// C/D 16x16 f32     : VGPR v, lanes 0-15 -> (M=v, N=lane);
//                     lanes 16-31 -> (M=v+8, N=lane-16).
// ---------------------------------------------------------------------------
__global__ __launch_bounds__(256) void zw_gemm_wmma(const float* __restrict__ Z,
                                                    const float* __restrict__ W,
                                                    float* __restrict__ U)
{
    const int lane = threadIdx.x & 31;
    const int wave = threadIdx.x >> 5;   // 8 waves -> 8 column tiles (N=128)
    const int m0   = blockIdx.x * 16;
    const int n0   = wave * 16;

    const int half = lane >> 4;          // 0 for lanes 0-15, 1 for lanes 16-31
    const int l15  = lane & 15;

    const float* zrow = Z + (size_t)(m0 + l15) * LATENT;
    const int    col  = n0 + l15;

    v8f c = {0.f, 0.f, 0.f, 0.f, 0.f, 0.f, 0.f, 0.f};

    #pragma unroll 4
    for (int k = 0; k < LATENT; k += 4) {
        const int kk = k + 2 * half;
        v2f a, b;
        a.x = zrow[kk];
        a.y = zrow[kk + 1];
        b.x = W[(size_t)kk * LATENT + col];
        b.y = W[(size_t)(kk + 1) * LATENT + col];
        c = __builtin_amdgcn_wmma_f32_16x16x4_f32(
                /*neg_a=*/false, a, /*neg_b=*/false, b,
                /*c_mod=*/(short)0, c, /*reuse_a=*/false, /*reuse_b=*/false);
    }

    float* urow = U + (size_t)(m0 + 8 * half) * LATENT + col;
    #pragma unroll
    for (int v = 0; v < 8; ++v)
        urow[(size_t)v * LATENT] = c[v];
}

// ---------------------------------------------------------------------------
// Kernel 2: per-edge  out[e] = sigmoid( U[j] . Z[i] )
// One wave32 per edge: lane loads float4 of each 512B row (fully coalesced),
// butterfly reduce across 32 lanes.
// ---------------------------------------------------------------------------
__global__ __launch_bounds__(256) void edge_dot_sigmoid(const float* __restrict__ Z,
                                                        const float* __restrict__ U,
                                                        const int* __restrict__ idx,
                                                        float* __restrict__ out,
                                                        int E)
{
    const int lane = threadIdx.x & 31;
    const int wave = threadIdx.x >> 5;
    const int e    = blockIdx.x * 8 + wave;
    if (e >= E) return;

    const int i = idx[e];        // row of z_i
    const int j = idx[E + e];    // row of z_j  (U[j] = z_j @ W)

    const float4 a = reinterpret_cast<const float4*>(U + (size_t)j * LATENT)[lane];
    const float4 b = reinterpret_cast<const float4*>(Z + (size_t)i * LATENT)[lane];

    float p = a.x * b.x + a.y * b.y + a.z * b.z + a.w * b.w;

    #pragma unroll
    for (int off = 16; off > 0; off >>= 1)
        p += __shfl_xor(p, off, 32);

    if (lane == 0)
        out[e] = 1.0f / (1.0f + expf(-p));
}

// ---------------------------------------------------------------------------
// Fallback (only if workspace is too small for U): per-edge matvec with
// L2-resident W. Deterministic: selection depends only on constant ws_size.
// ---------------------------------------------------------------------------
__global__ __launch_bounds__(256) void edge_direct(const float* __restrict__ Z,
                                                   const float* __restrict__ W,
                                                   const int* __restrict__ idx,
                                                   float* __restrict__ out,
                                                   int E)
{
    const int lane = threadIdx.x & 31;
    const int wave = threadIdx.x >> 5;
    const int e    = blockIdx.x * 8 + wave;
    if (e >= E) return;

    const int i = idx[e];
    const int j = idx[E + e];

    const float* zi = Z + (size_t)i * LATENT;
    const float* zj = Z + (size_t)j * LATENT;

    float4 acc = {0.f, 0.f, 0.f, 0.f};
    for (int d = 0; d < LATENT; ++d) {
        const float  zd = zj[d];                      // wave-broadcast load
        const float4 w4 = reinterpret_cast<const float4*>(W + (size_t)d * LATENT)[lane];
        acc.x += zd * w4.x;
        acc.y += zd * w4.y;
        acc.z += zd * w4.z;
        acc.w += zd * w4.w;
    }

    const float4 b = reinterpret_cast<const float4*>(zi)[lane];
    float p = acc.x * b.x + acc.y * b.y + acc.z * b.z + acc.w * b.w;

    #pragma unroll
    for (int off = 16; off > 0; off >>= 1)
        p += __shfl_xor(p, off, 32);

    if (lane == 0)
        out[e] = 1.0f / (1.0f + expf(-p));
}

extern "C" void kernel_launch(void* const* d_in, const int* in_sizes, int n_in,
                              void* d_out, int out_size, void* d_ws, size_t ws_size,
                              hipStream_t stream)
{
    const float* Z   = (const float*)d_in[0];   // [N, 128]
    const int*   idx = (const int*)d_in[1];     // [2, E] flat
    const float* W   = (const float*)d_in[2];   // [128, 128]
    float*       out = (float*)d_out;           // [E]

    const int N = in_sizes[0] / LATENT;         // 100000
    const int E = in_sizes[1] / 2;              // 600000

    const size_t needU = (size_t)N * LATENT * sizeof(float);   // 51.2 MB

    if (ws_size >= needU) {
        float* U = (float*)d_ws;
        zw_gemm_wmma<<<dim3(N / 16), dim3(256), 0, stream>>>(Z, W, U);
        edge_dot_sigmoid<<<dim3((E + 7) / 8), dim3(256), 0, stream>>>(Z, U, idx, out, E);
    } else {
        edge_direct<<<dim3((E + 7) / 8), dim3(256), 0, stream>>>(Z, W, idx, out, E);
    }
}